// GasSplattingModel_21998822490245
// MI455X (gfx1250) — compile-verified
//
#include <hip/hip_runtime.h>
#include <hip/hip_bf16.h>
#include <math.h>

#define EPS_F    1e-7f
#define K_GAUSS  4096
#define R_RAYS   8192

// k = -0.5*log2(e);  s = sqrt(-k);  exp(-0.5*x) == exp2(k*x)
#define KEXP_F  (-0.72134752044448170368f)
#define SEXP_F  ( 0.84932180028801904272f)
#define SQRT_2PI_F (2.50662827463100050242f)

typedef float v2f __attribute__((ext_vector_type(2)));
typedef float v8f __attribute__((ext_vector_type(8)));

// ---------------------------------------------------------------------------
// Kernel 1: per-Gaussian preprocessing (K = 4096).
// Workspace is written directly in WMMA B-operand layout, split per lane
// half (lo = lanes 0-15 hold K0,K1 / K4,K5; hi = lanes 16-31 hold K2,K3 /
// K6,K7), 24 floats (6 float4) per Gaussian:
//
//   contractions:
//     a   = qu  . [m00, m01, m11]                       (K=3)
//     tb  = s * ( qup . M  -  u . Sm )                  (K=5, scaled by s)
//     kc0 = k * ( qp . M  - 2 p . Sm  + mSm )           (K=6, scaled by k)
//
//   lo block (float4 x3):
//     { m00,       m01,    s*m00, s*m01 }    // B_a, B_b1
//     { -s*Sm1,    concs,  k*m00, k*m01 }    // B_b2 (concs rides dead K5 slot), B_c1
//     { -2k*Sm1,   k*mSm,  0,     0     }    // B_c2
//   hi block (float4 x3):
//     { m11,       0,      s*m11, -s*Sm0 }
//     { 0,         concs,  k*m11, -2k*Sm0 }
//     { 0,         0,      0,     0      }
// ---------------------------------------------------------------------------
__global__ __launch_bounds__(256)
void gas_prep_kernel(const float* __restrict__ pos_raw,
                     const float* __restrict__ conc_raw,
                     const float* __restrict__ scale_raw,
                     const float* __restrict__ rot_raw,
                     const int*   __restrict__ map_size,
                     float* __restrict__ W)
{
    int k = blockIdx.x * blockDim.x + threadIdx.x;
    if (k >= K_GAUSS) return;

    float ms = (float)map_size[0];

    float p0 = ms / (1.0f + expf(-pos_raw[2 * k + 0]));
    float p1 = ms / (1.0f + expf(-pos_raw[2 * k + 1]));

    float cx    = conc_raw[k];
    float conc  = fmaxf(cx, 0.0f) + log1pf(expf(-fabsf(cx)));
    float concs = conc * SQRT_2PI_F;

    float s0 = expf(scale_raw[2 * k + 0]);
    float s1 = expf(scale_raw[2 * k + 1]);
    float d0 = 1.0f / (s0 * s0 + EPS_F);
    float d1 = 1.0f / (s1 * s1 + EPS_F);

    float rr = rot_raw[k];
    float cr = cosf(rr);
    float sr = sinf(rr);

    float m00 = cr * cr * d0 + sr * sr * d1;
    float m01 = cr * sr * (d0 - d1);
    float m11 = sr * sr * d0 + cr * cr * d1;

    float Sm0 = m00 * p0 + m01 * p1;
    float Sm1 = m01 * p0 + m11 * p1;
    float mSm = p0 * Sm0 + p1 * Sm1;

    const float kk = KEXP_F;
    const float ss = SEXP_F;

    float4* W4 = (float4*)W;
    // lo-half block (lanes 0-15)
    W4[6 * k + 0] = make_float4(m00, m01, ss * m00, ss * m01);
    W4[6 * k + 1] = make_float4(-ss * Sm1, concs, kk * m00, kk * m01);
    W4[6 * k + 2] = make_float4(-2.0f * kk * Sm1, kk * mSm, 0.0f, 0.0f);
    // hi-half block (lanes 16-31)
    W4[6 * k + 3] = make_float4(m11, 0.0f, ss * m11, -ss * Sm0);
    W4[6 * k + 4] = make_float4(0.0f, concs, kk * m11, -2.0f * kk * Sm0);
    W4[6 * k + 5] = make_float4(0.0f, 0.0f, 0.0f, 0.0f);
}

// ---------------------------------------------------------------------------
// Kernel 2: one wave32 per 16-ray tile; 256 tiles of 16 Gaussians.
// Per tile: 3 x global_load_b128, 5 x V_WMMA_F32_16X16X4_F32
// (a:1, tb:2 chained, kc0:2 chained), then per-element tail
// v_rsq + v_exp + pk_mul/pk_fma. Zero cndmasks in the loop.
// A 16x4 f32 layout: lane m (0..15) holds {K0,K1}; lanes 16..31 hold {K2,K3}.
// C/D: VGPR v, lane l -> row v + 8*(l>=16), col l%16.
// ---------------------------------------------------------------------------
__global__ __launch_bounds__(256)
void gas_splat_kernel(const float* __restrict__ p_rays,
                      const float* __restrict__ u_rays,
                      const float* __restrict__ W,
                      float* __restrict__ out)
{
    const int lane = threadIdx.x & 31;
    const int wid  = threadIdx.x >> 5;
    const int tile = blockIdx.x * 8 + wid;
    const int mrow = lane & 15;
    const bool hi  = lane >= 16;
    const int ray  = tile * 16 + mrow;

    float u0 = u_rays[2 * ray + 0];
    float u1 = u_rays[2 * ray + 1];
    float p0 = p_rays[2 * ray + 0];
    float p1 = p_rays[2 * ray + 1];

    // Loop-invariant A operands (K rows of the three contractions):
    // a  : [u0^2, 2u0u1, u1^2, 0]
    // tb : [u0p0, u0p1+u1p0, u1p1, u0 | u1, 0, 0, 0]
    // kc0: [p0^2, 2p0p1, p1^2, p0   | p1, 1, 0, 0]
    v2f A_a, A_b1, A_b2, A_c1, A_c2;
    A_a.x  = hi ? (u1 * u1) : (u0 * u0);
    A_a.y  = hi ? 0.0f      : (2.0f * u0 * u1);
    A_b1.x = hi ? (u1 * p1) : (u0 * p0);
    A_b1.y = hi ? u0        : (u0 * p1 + u1 * p0);
    A_b2.x = hi ? 0.0f      : u1;
    A_b2.y = 0.0f;
    A_c1.x = hi ? (p1 * p1) : (p0 * p0);
    A_c1.y = hi ? p0        : (2.0f * p0 * p1);
    A_c2.x = hi ? 0.0f      : p1;
    A_c2.y = hi ? 0.0f      : 1.0f;

    v8f acc = {};
    const float4* G = (const float4*)W;
    const int half_off = hi ? 3 : 0;   // float4 offset into per-Gaussian block

    for (int kb = 0; kb < K_GAUSS; kb += 16) {
        const int base = 6 * (kb + mrow) + half_off;
        float4 g0 = G[base + 0];
        float4 g1 = G[base + 1];
        float4 g2 = G[base + 2];

        v2f B_a, B_b1, B_b2, B_c1, B_c2;
        B_a.x  = g0.x;  B_a.y  = g0.y;
        B_b1.x = g0.z;  B_b1.y = g0.w;
        B_b2.x = g1.x;  B_b2.y = g1.y;   // .y = concs (A slot is 0 there)
        B_c1.x = g1.z;  B_c1.y = g1.w;
        B_c2.x = g2.x;  B_c2.y = g2.y;

        float concs = g1.y;              // conc * sqrt(2pi) for column n

        v8f zz = {};
        v8f a_t = __builtin_amdgcn_wmma_f32_16x16x4_f32(
                      false, A_a, false, B_a, (short)0, zz, false, false);
        v8f b_t = __builtin_amdgcn_wmma_f32_16x16x4_f32(
                      false, A_b1, false, B_b1, (short)0, zz, false, false);
        b_t     = __builtin_amdgcn_wmma_f32_16x16x4_f32(
                      false, A_b2, false, B_b2, (short)0, b_t, false, false);
        v8f c_t = __builtin_amdgcn_wmma_f32_16x16x4_f32(
                      false, A_c1, false, B_c1, (short)0, zz, false, false);
        c_t     = __builtin_amdgcn_wmma_f32_16x16x4_f32(
                      false, A_c2, false, B_c2, (short)0, c_t, false, false);

        // dens = concs * rsq(a) * exp2( kc0 + (s*b*rsq(a))^2 )
        #pragma unroll
        for (int v = 0; v < 8; ++v) {
            float rs  = __builtin_amdgcn_rsqf(a_t[v]);
            float tb  = b_t[v] * rs;
            float arg = fmaf(tb, tb, c_t[v]);
            float e   = __builtin_amdgcn_exp2f(arg);
            acc[v]    = fmaf(concs * rs, e, acc[v]);
        }
    }

    // Reduce over the 16 Gaussian columns held by each 16-lane half-wave.
    #pragma unroll
    for (int off = 1; off < 16; off <<= 1) {
        #pragma unroll
        for (int v = 0; v < 8; ++v)
            acc[v] += __shfl_xor(acc[v], off, 32);
    }

    // lane 0 holds rows 0..7 of the tile, lane 16 holds rows 8..15.
    if (mrow == 0) {
        int base = tile * 16 + (hi ? 8 : 0);
        #pragma unroll
        for (int v = 0; v < 8; ++v)
            out[base + v] = acc[v];
    }
}

// ---------------------------------------------------------------------------
extern "C" void kernel_launch(void* const* d_in, const int* in_sizes, int n_in,
                              void* d_out, int out_size, void* d_ws, size_t ws_size,
                              hipStream_t stream)
{
    const float* pos_raw   = (const float*)d_in[0];
    const float* conc_raw  = (const float*)d_in[1];
    const float* scale_raw = (const float*)d_in[2];
    const float* rot_raw   = (const float*)d_in[3];
    const float* p_rays    = (const float*)d_in[4];
    const float* u_rays    = (const float*)d_in[5];
    const int*   map_size  = (const int*)d_in[6];

    float* W   = (float*)d_ws;   // 4096 * 24 floats = 384 KB
    float* out = (float*)d_out;

    gas_prep_kernel<<<K_GAUSS / 256, 256, 0, stream>>>(
        pos_raw, conc_raw, scale_raw, rot_raw, map_size, W);

    // 512 ray-tiles, 8 waves per 256-thread block -> 64 blocks.
    gas_splat_kernel<<<R_RAYS / 16 / 8, 256, 0, stream>>>(
        p_rays, u_rays, W, out);
}